// KPConvEncoder_7112465842599
// MI455X (gfx1250) — compile-verified
//
#include <hip/hip_runtime.h>
#include <hip/hip_bf16.h>

#define BB 8
#define NN 4096
#define KNB 16
#define PP 15
#define ZD 256
#define R1 0.1f
#define R2 0.2f

typedef float v2f __attribute__((ext_vector_type(2)));
typedef float v8f __attribute__((ext_vector_type(8)));

// ---------------------------------------------------------------------------
// Kernel A: fused kNN (no materialized distance matrix).
// One thread per query point; candidates staged through LDS in 256-pt tiles;
// top-16 kept sorted in registers (fully unrolled, static indices only).
// ---------------------------------------------------------------------------
__global__ void __launch_bounds__(256) knn_kernel(const float* __restrict__ x,
                                                  int* __restrict__ idxout) {
  __shared__ float tile[256 * 3];
  const int gid = blockIdx.x * 256 + threadIdx.x;
  const int b = gid / NN;          // 16 blocks per batch -> block never straddles batches
  const int n = gid % NN;
  const float* xb = x + (size_t)b * NN * 3;
  const float qx = xb[n * 3 + 0];
  const float qy = xb[n * 3 + 1];
  const float qz = xb[n * 3 + 2];

  float bd[KNB];
  int bi[KNB];
#pragma unroll
  for (int t = 0; t < KNB; ++t) { bd[t] = 3.4e38f; bi[t] = 0; }

  for (int t0 = 0; t0 < NN; t0 += 256) {
    __syncthreads();
    const int jl = t0 + threadIdx.x;
    tile[threadIdx.x * 3 + 0] = xb[jl * 3 + 0];
    tile[threadIdx.x * 3 + 1] = xb[jl * 3 + 1];
    tile[threadIdx.x * 3 + 2] = xb[jl * 3 + 2];
    __syncthreads();
    for (int jj = 0; jj < 256; ++jj) {
      const float dx = tile[jj * 3 + 0] - qx;
      const float dy = tile[jj * 3 + 1] - qy;
      const float dz = tile[jj * 3 + 2] - qz;
      const float d2 = dx * dx + dy * dy + dz * dz;
      if (d2 < bd[KNB - 1]) {            // strict <: stable tie-break by index
        bd[KNB - 1] = d2;
        bi[KNB - 1] = t0 + jj;
#pragma unroll
        for (int t = KNB - 1; t > 0; --t) {
          if (bd[t] < bd[t - 1]) {
            float td = bd[t]; bd[t] = bd[t - 1]; bd[t - 1] = td;
            int ti = bi[t];   bi[t] = bi[t - 1]; bi[t - 1] = ti;
          }
        }
      }
    }
  }
  int* op = idxout + ((size_t)b * NN + n) * KNB;
#pragma unroll
  for (int t = 0; t < KNB; ++t) op[t] = bi[t];
}

// ---------------------------------------------------------------------------
// Kernel B: KPConv layer 1 (+ReLU). One thread per point. W1/kp1 in LDS.
// ---------------------------------------------------------------------------
__global__ void __launch_bounds__(256) kpconv1_kernel(const float* __restrict__ x,
                                                      const int* __restrict__ idx,
                                                      const float* __restrict__ kp1,
                                                      const float* __restrict__ W1,
                                                      float* __restrict__ f1) {
  __shared__ float W1s[PP * 3 * 32];
  __shared__ float kps[PP * 3];
  for (int i = threadIdx.x; i < PP * 3 * 32; i += 256) W1s[i] = W1[i];
  if (threadIdx.x < PP * 3) kps[threadIdx.x] = kp1[threadIdx.x];
  __syncthreads();

  const int gid = blockIdx.x * 256 + threadIdx.x;
  const int b = gid / NN;
  const int n = gid % NN;
  const float* xb = x + (size_t)b * NN * 3;
  const float qx = xb[n * 3 + 0], qy = xb[n * 3 + 1], qz = xb[n * 3 + 2];
  const int* ip = idx + ((size_t)b * NN + n) * KNB;

  float agg[PP][3];
#pragma unroll
  for (int p = 0; p < PP; ++p) { agg[p][0] = 0.f; agg[p][1] = 0.f; agg[p][2] = 0.f; }

  const float invr = 1.0f / R1;
  for (int k = 0; k < KNB; ++k) {
    const int j = ip[k];
    const float nx = xb[j * 3 + 0], ny = xb[j * 3 + 1], nz = xb[j * 3 + 2];
    const float dx = nx - qx, dy = ny - qy, dz = nz - qz;
    const float dn = sqrtf(dx * dx + dy * dy + dz * dz);
    if (dn <= R1) {                       // in_ball mask (zero contribution otherwise)
#pragma unroll
      for (int p = 0; p < PP; ++p) {
        const float ex = dx - kps[p * 3 + 0];
        const float ey = dy - kps[p * 3 + 1];
        const float ez = dz - kps[p * 3 + 2];
        const float dkp = sqrtf(ex * ex + ey * ey + ez * ez);
        const float w = fmaxf(1.0f - dkp * invr, 0.0f);
        agg[p][0] += w * nx; agg[p][1] += w * ny; agg[p][2] += w * nz;
      }
    }
  }

  float* outp = f1 + ((size_t)b * NN + n) * 32;
  for (int d = 0; d < 32; ++d) {
    float s = 0.0f;
#pragma unroll
    for (int p = 0; p < PP; ++p) {
      s += agg[p][0] * W1s[p * 96 + 0 * 32 + d];
      s += agg[p][1] * W1s[p * 96 + 1 * 32 + d];
      s += agg[p][2] * W1s[p * 96 + 2 * 32 + d];
    }
    outp[d] = fmaxf(s, 0.0f);            // ReLU
  }
}

__global__ void init_g_kernel(int* __restrict__ g) {
  if (threadIdx.x < BB * 64) g[threadIdx.x] = 0;
}

// ---------------------------------------------------------------------------
// Kernel C: KPConv layer 2, fused: influence -> agg tile (LDS) ->
// f32 WMMA GEMM [16 x 480] @ [480 x 64] -> ReLU -> per-wave max -> atomicMax.
// One workgroup = 16 points, 4 wave32's (each owns a 16-wide N-tile).
// ---------------------------------------------------------------------------
__global__ void __launch_bounds__(128) kpconv2_kernel(const float* __restrict__ x,
                                                      const int* __restrict__ idx,
                                                      const float* __restrict__ kp2,
                                                      const float* __restrict__ f1,
                                                      const float* __restrict__ W2,
                                                      int* __restrict__ gmax) {
  __shared__ float inflS[16 * 16 * 16];   // [pt][k][p] (p padded 15->16)  16 KB
  __shared__ float aggS[16 * 480];        // [pt][p*32+c]                  30 KB
  const int tid = threadIdx.x;
  const int b = blockIdx.x / (NN / 16);
  const int n_base = (blockIdx.x % (NN / 16)) * 16;
  const float* xb = x + (size_t)b * NN * 3;
  const int* idxb = idx + (size_t)b * NN * KNB;
  const float invr = 1.0f / R2;

  // ---- phase 1: influences, 2 (pt,k) pairs per thread ----
  {
    const int pid = tid >> 3;
    const int kk = (tid & 7) * 2;
    const int n = n_base + pid;
    const float qx = xb[n * 3 + 0], qy = xb[n * 3 + 1], qz = xb[n * 3 + 2];
    for (int ko = 0; ko < 2; ++ko) {
      const int k = kk + ko;
      const int j = idxb[n * KNB + k];
      const float dx = xb[j * 3 + 0] - qx;
      const float dy = xb[j * 3 + 1] - qy;
      const float dz = xb[j * 3 + 2] - qz;
      const float dn = sqrtf(dx * dx + dy * dy + dz * dz);
      const float ib = (dn <= R2) ? 1.0f : 0.0f;
#pragma unroll
      for (int p = 0; p < PP; ++p) {
        const float ex = dx - kp2[p * 3 + 0];
        const float ey = dy - kp2[p * 3 + 1];
        const float ez = dz - kp2[p * 3 + 2];
        const float dkp = sqrtf(ex * ex + ey * ey + ez * ez);
        inflS[pid * 256 + k * 16 + p] = fmaxf(1.0f - dkp * invr, 0.0f) * ib;
      }
    }
  }
  __syncthreads();

  // ---- phase 2: agg[pt][p][c] = sum_k infl[pt][k][p]*f1[idx[k]][c];
  //      thread owns (pt, 4 channels) ----
  {
    const int pid = tid >> 3;
    const int c0 = (tid & 7) * 4;
    const int n = n_base + pid;
    float acc[PP][4];
#pragma unroll
    for (int p = 0; p < PP; ++p) {
      acc[p][0] = 0.f; acc[p][1] = 0.f; acc[p][2] = 0.f; acc[p][3] = 0.f;
    }
    for (int k = 0; k < KNB; ++k) {
      const int j = idxb[n * KNB + k];
      const float* fp = f1 + ((size_t)b * NN + j) * 32 + c0;
      const float g0 = fp[0], g1 = fp[1], g2 = fp[2], g3 = fp[3];
#pragma unroll
      for (int p = 0; p < PP; ++p) {
        const float w = inflS[pid * 256 + k * 16 + p];
        acc[p][0] += w * g0; acc[p][1] += w * g1;
        acc[p][2] += w * g2; acc[p][3] += w * g3;
      }
    }
#pragma unroll
    for (int p = 0; p < PP; ++p) {
      aggS[pid * 480 + p * 32 + c0 + 0] = acc[p][0];
      aggS[pid * 480 + p * 32 + c0 + 1] = acc[p][1];
      aggS[pid * 480 + p * 32 + c0 + 2] = acc[p][2];
      aggS[pid * 480 + p * 32 + c0 + 3] = acc[p][3];
    }
  }
  __syncthreads();

  // ---- phase 3: exact-f32 WMMA GEMM, 120 x v_wmma_f32_16x16x4_f32 ----
  {
    const int wave = tid >> 5;
    const int lane = tid & 31;
    const int M = lane & 15;            // A row / B col / C col
    const int hi = lane >> 4;           // K-half selector for A/B, row-half for C
    const int n0 = wave * 16;           // this wave's output-column tile
    v8f acc = {0.f, 0.f, 0.f, 0.f, 0.f, 0.f, 0.f, 0.f};
    const float* arow = aggS + M * 480 + 2 * hi;                 // A: 16x4 tiles
    const float* bcol = W2 + (size_t)(2 * hi) * 64 + n0 + M;     // B: 4x16 tiles (K stride 64)
    for (int kb = 0; kb < 480; kb += 4) {
      __builtin_prefetch(bcol + (size_t)(kb + 8) * 64, 0, 0);    // global_prefetch_b8
      v2f av, bv;
      av.x = arow[kb];
      av.y = arow[kb + 1];
      bv.x = bcol[(size_t)kb * 64];
      bv.y = bcol[(size_t)(kb + 1) * 64];
      acc = __builtin_amdgcn_wmma_f32_16x16x4_f32(
          /*neg_a=*/false, av, /*neg_b=*/false, bv,
          /*c_mod=*/(short)0, acc, /*reuse_a=*/false, /*reuse_b=*/false);
    }
    // ReLU + max over the 16 rows this wave holds for column d = n0 + M:
    // lane holds rows v + 8*hi (v=0..7); partner lane (lane^16) holds the other half.
    float m = 0.0f;                      // >= 0 == ReLU folded into max
#pragma unroll
    for (int v = 0; v < 8; ++v) m = fmaxf(m, acc[v]);
    m = fmaxf(m, __shfl_xor(m, 16, 32));
    if (lane < 16) {
      // non-negative floats: int compare == float compare
      atomicMax(&gmax[b * 64 + n0 + lane], __float_as_int(m));
    }
  }
}

// ---------------------------------------------------------------------------
// Kernel D: g(8x64) @ fc_w(64x512) + fc_b -> (mu || sigma) tuple layout.
// ---------------------------------------------------------------------------
__global__ void __launch_bounds__(256) fc_kernel(const int* __restrict__ gmax,
                                                 const float* __restrict__ fcw,
                                                 const float* __restrict__ fcb,
                                                 float* __restrict__ out) {
  const int gid = blockIdx.x * 256 + threadIdx.x;
  if (gid >= BB * 2 * ZD) return;
  const int b = gid >> 9;
  const int j = gid & 511;
  float s = fcb[j];
  for (int c = 0; c < 64; ++c)
    s += __int_as_float(gmax[b * 64 + c]) * fcw[c * (2 * ZD) + j];
  if (j < ZD) out[b * ZD + j] = s;                       // mu
  else        out[BB * ZD + b * ZD + (j - ZD)] = s;      // sigma
}

// ---------------------------------------------------------------------------
extern "C" void kernel_launch(void* const* d_in, const int* in_sizes, int n_in,
                              void* d_out, int out_size, void* d_ws, size_t ws_size,
                              hipStream_t stream) {
  const float* x   = (const float*)d_in[0];
  const float* kp1 = (const float*)d_in[1];
  const float* W1  = (const float*)d_in[2];
  const float* kp2 = (const float*)d_in[3];
  const float* W2  = (const float*)d_in[4];
  const float* fcw = (const float*)d_in[5];
  const float* fcb = (const float*)d_in[6];
  float* out = (float*)d_out;

  char* ws = (char*)d_ws;
  int*   idxbuf = (int*)ws;                                            // 2 MB
  float* f1buf  = (float*)(ws + (size_t)BB * NN * KNB * 4);            // 4 MB
  int*   gbuf   = (int*)(ws + (size_t)BB * NN * KNB * 4
                            + (size_t)BB * NN * 32 * 4);               // 2 KB

  knn_kernel<<<BB * NN / 256, 256, 0, stream>>>(x, idxbuf);
  kpconv1_kernel<<<BB * NN / 256, 256, 0, stream>>>(x, idxbuf, kp1, W1, f1buf);
  init_g_kernel<<<1, 512, 0, stream>>>(gbuf);
  kpconv2_kernel<<<BB * NN / 16, 128, 0, stream>>>(x, idxbuf, kp2, f1buf, W2, gbuf);
  fc_kernel<<<(BB * 2 * ZD + 255) / 256, 256, 0, stream>>>(gbuf, fcw, fcb, out);
}